// RetinaATSSNetTrafficLoss_78709570667225
// MI455X (gfx1250) — compile-verified
//
#include <hip/hip_runtime.h>
#include <math.h>

// ---------------- problem constants (match reference) ----------------
#define N_IMG   32
#define N_OBJ   16
#define N_LEV   5
#define NCAND   11
#define NCLS    20
#define P_TOT   21824
#define GAMMA_F 2.5f
#define ALPHA_F 0.25f

// level tables: W (=H), obj scale, anchor offset
__device__ __constant__ int   c_W[N_LEV]   = {128, 64, 32, 16, 8};
__device__ __constant__ float c_S[N_LEV]   = {0.1f, 0.2f, 0.4f, 0.6f, 0.8f};
__device__ __constant__ int   c_OFF[N_LEV] = {0, 16384, 20480, 21504, 21760};

// ---------------- workspace layout ----------------
// bytes [0..15]   : float conf_acc, float loc_acc, float pad, unsigned n_pos
// bytes [16..143] : int counts[32]
// bytes [160.. ]  : Ent entries, 64 per image, 24B each (32*64*24 = 49152 B)
struct Ent { int p; int lab; float b0, b1, b2, b3; };   // 24 bytes
#define WS_ENT_OFF 160
#define ENT_PER_IMG 64

// ---------------- WMMA wave reduction ----------------
typedef __attribute__((ext_vector_type(2))) float v2f;
typedef __attribute__((ext_vector_type(8))) float v8f;

// Sum one float per lane across the wave using the matrix pipe:
// A (16x4 f32): VGPR0 = lane partial (K=0 for lanes 0-15, K=2 for 16-31), VGPR1 = 0.
// B (4x16 f32) = all ones  =>  D[m][n] = p_m + p_{m+16}  (replicated over n).
// C/D lanes hold D[r][n] (r=0..7) / D[r+8][n]; summing the 8 VGPRs and one
// shfl_xor(16) gives every lane the full wave sum. Requires EXEC == all ones.
__device__ __forceinline__ float wave_sum_wmma(float v) {
    v2f a; a.x = v;   a.y = 0.0f;
    v2f b; b.x = 1.0f; b.y = 1.0f;
    v8f c = {};
    c = __builtin_amdgcn_wmma_f32_16x16x4_f32(false, a, false, b, (short)0, c,
                                              false, false);
    float s = c[0] + c[1] + c[2] + c[3] + c[4] + c[5] + c[6] + c[7];
    s += __shfl_xor(s, 16, 32);
    return s;
}

// ---------------- focal helpers ----------------
// log_sigmoid(x) = min(x,0) - log1p(exp(-|x|)); both signs share log1p term.
__device__ __forceinline__ float focal_neg(float x) {
    float ax   = fabsf(x);
    float l1p  = log1pf(__expf(-ax));
    float ls_p = fminf(x, 0.0f) - l1p;    // log sigmoid(x)
    float ls_n = fminf(-x, 0.0f) - l1p;   // log sigmoid(-x)
    float pg   = __expf(GAMMA_F * ls_p);  // sigmoid(x)^gamma
    return -(1.0f - ALPHA_F) * pg * ls_n; // all-negative (match=0) term
}

__device__ __forceinline__ float smooth_l1(float d) {
    d = fabsf(d);
    return (d < 1.0f) ? 0.5f * d * d : d - 0.5f;
}

// ---------------- K0: init accumulators ----------------
__global__ void k_init(char* ws) {
    if (threadIdx.x < 40) ((unsigned*)ws)[threadIdx.x] = 0u;  // acc + counts
}

// ---------------- K1: ATSS assignment (1 block / image, 1 wave / object) ----
__global__ __launch_bounds__(512) void k_assign(
    const float* __restrict__ boxes,   // (N,16,4) xyxy
    const int*   __restrict__ labels,  // (N,16)
    char* ws) {
    __shared__ float s_ov [N_OBJ][N_LEV * NCAND];
    __shared__ int   s_idx[N_OBJ][N_LEV * NCAND];   // global anchor index
    __shared__ int   s_in [N_OBJ][N_LEV * NCAND];
    __shared__ float s_thr[N_OBJ];
    __shared__ int   s_cnt;
    __shared__ Ent   s_ent[ENT_PER_IMG];

    const int img  = blockIdx.x;
    const int lane = threadIdx.x & 31;
    const int ob   = threadIdx.x >> 5;      // wave id == object id (16 waves)

    if (threadIdx.x == 0) s_cnt = 0;

    const float* bb = boxes + (img * N_OBJ + ob) * 4;
    const float bx0 = bb[0], by0 = bb[1], bx1 = bb[2], by1 = bb[3];
    const float bcx = 0.5f * (bx0 + bx1), bcy = 0.5f * (by0 + by1);
    const float barea = (bx1 - bx0) * (by1 - by0);

    for (int l = 0; l < N_LEV; ++l) {
        const int   W = c_W[l];
        const float s = c_S[l];
        const int   Pl = W * W;
        const float invW = 1.0f / (float)W;

        // per-lane top-11 by (d^2, idx) ascending, register resident
        float kd[NCAND]; int ki[NCAND];
#pragma unroll
        for (int q = 0; q < NCAND; ++q) { kd[q] = INFINITY; ki[q] = 0x7fffffff; }

        for (int j = lane; j < Pl; j += 32) {
            int y = j / W, x = j - y * W;
            float cx = ((float)x + 0.5f) * invW;
            float cy = ((float)y + 0.5f) * invW;
            float dx = bcx - cx, dy = bcy - cy;
            float d2 = dx * dx + dy * dy;
            bool better = (d2 < kd[NCAND - 1]) ||
                          (d2 == kd[NCAND - 1] && j < ki[NCAND - 1]);
            if (better) {
                kd[NCAND - 1] = d2; ki[NCAND - 1] = j;
#pragma unroll
                for (int q = NCAND - 1; q > 0; --q) {
                    bool sw = (kd[q] < kd[q - 1]) ||
                              (kd[q] == kd[q - 1] && ki[q] < ki[q - 1]);
                    if (sw) {
                        float td = kd[q]; kd[q] = kd[q - 1]; kd[q - 1] = td;
                        int   ti = ki[q]; ki[q] = ki[q - 1]; ki[q - 1] = ti;
                    }
                }
            }
        }

        // 11 rounds of wave arg-min -> candidates emitted in top_k order
        for (int cslot = 0; cslot < NCAND; ++cslot) {
            float md = INFINITY; int mi = 0x7fffffff;
#pragma unroll
            for (int q = 0; q < NCAND; ++q) {
                bool lt = (kd[q] < md) || (kd[q] == md && ki[q] < mi);
                if (lt) { md = kd[q]; mi = ki[q]; }
            }
#pragma unroll
            for (int off = 16; off >= 1; off >>= 1) {
                float od = __shfl_xor(md, off, 32);
                int   oi = __shfl_xor(mi, off, 32);
                bool lt = (od < md) || (od == md && oi < mi);
                if (lt) { md = od; mi = oi; }
            }
            // owner retires the winner (anchor indices are unique)
#pragma unroll
            for (int q = 0; q < NCAND; ++q)
                if (ki[q] == mi) { kd[q] = INFINITY; ki[q] = 0x7fffffff; }

            if (lane == 0) {
                int y = mi / W, x = mi - y * W;
                float cx = ((float)x + 0.5f) * invW;
                float cy = ((float)y + 0.5f) * invW;
                float half = 0.5f * s;
                float ix0 = fmaxf(bx0, cx - half), iy0 = fmaxf(by0, cy - half);
                float ix1 = fminf(bx1, cx + half), iy1 = fminf(by1, cy + half);
                float iw = fmaxf(ix1 - ix0, 0.0f), ih = fmaxf(iy1 - iy0, 0.0f);
                float inter = iw * ih;
                float ov = inter / (barea + s * s - inter);
                bool ins = (bx0 <= cx) && (cx <= bx1) && (by0 <= cy) && (cy <= by1);
                int slot = l * NCAND + cslot;
                s_ov[ob][slot]  = ov;
                s_idx[ob][slot] = c_OFF[l] + mi;
                s_in[ob][slot]  = ins ? 1 : 0;
            }
        }
    }

    // thr = mean + std(ddof=1) over the 55 candidate IoUs (per object/wave).
    // Block is fully converged here -> EXEC all ones -> WMMA reduction legal.
    {
        float sum = 0.0f;
        for (int t = lane; t < N_LEV * NCAND; t += 32) sum += s_ov[ob][t];
        sum = wave_sum_wmma(sum);
        float mean = sum * (1.0f / 55.0f);
        float ss = 0.0f;
        for (int t = lane; t < N_LEV * NCAND; t += 32) {
            float d = s_ov[ob][t] - mean; ss += d * d;
        }
        ss = wave_sum_wmma(ss);
        if (lane == 0) s_thr[ob] = mean + sqrtf(ss * (1.0f / 54.0f));
    }
    __syncthreads();

    // sequential slot scan, one thread per level (labels are per-level arrays)
    if (threadIdx.x < N_LEV) {
        int l = threadIdx.x;
        int assigned[NCAND]; int na = 0;
        for (int cslot = 0; cslot < NCAND; ++cslot) {
            float best = -1.0f; int bob = -1;
            int slot = l * NCAND + cslot;
            for (int o = 0; o < N_OBJ; ++o) {
                float ov = s_ov[o][slot];
                if (s_in[o][slot] && ov > s_thr[o] && ov > best) {
                    best = ov; bob = o;   // strict '>' keeps first arg-max
                }
            }
            if (bob >= 0) {
                int p = s_idx[bob][slot];
                bool dup = false;
                for (int q = 0; q < na; ++q) if (assigned[q] == p) dup = true;
                if (!dup) {                      // label[p]==0 check
                    assigned[na++] = p;
                    int e = atomicAdd(&s_cnt, 1);
                    const float* b2 = boxes + (img * N_OBJ + bob) * 4;
                    s_ent[e].p = p;
                    s_ent[e].lab = labels[img * N_OBJ + bob];
                    s_ent[e].b0 = b2[0]; s_ent[e].b1 = b2[1];
                    s_ent[e].b2 = b2[2]; s_ent[e].b3 = b2[3];
                }
            }
        }
    }
    __syncthreads();

    int cnt = s_cnt;
    int* counts = (int*)(ws + 16);
    Ent* ents = (Ent*)(ws + WS_ENT_OFF);
    if (threadIdx.x == 0) {
        counts[img] = cnt;
        atomicAdd((unsigned*)ws + 3, (unsigned)cnt);   // n_pos
    }
    for (int t = threadIdx.x; t < cnt; t += blockDim.x)
        ents[img * ENT_PER_IMG + t] = s_ent[t];
}

// ---------------- K2: streaming focal base (all-negative), WMMA reduce -----
__global__ __launch_bounds__(256) void k_focal_base(
    const float* __restrict__ scores, char* ws, long total4) {
    __shared__ float sh[8];
    const float4* s4 = (const float4*)scores;
    long i0 = (long)blockIdx.x * blockDim.x + threadIdx.x;
    long stride = (long)gridDim.x * blockDim.x;
    float local = 0.0f;
    for (long i = i0; i < total4; i += stride) {
        __builtin_prefetch(&s4[i + stride], 0, 1);   // global_prefetch_b8
        float4 v = s4[i];
        local += focal_neg(v.x) + focal_neg(v.y) +
                 focal_neg(v.z) + focal_neg(v.w);
    }
    float wsum = wave_sum_wmma(local);               // matrix-pipe reduction
    int wid = threadIdx.x >> 5;
    if ((threadIdx.x & 31) == 0) sh[wid] = wsum;
    __syncthreads();
    if (threadIdx.x == 0) {
        float t = 0.0f;
#pragma unroll
        for (int w = 0; w < 8; ++w) t += sh[w];
        atomicAdd((float*)ws, t);                    // conf accumulator
    }
}

// ---------------- K3: sparse positives: SmoothL1 + focal correction --------
__global__ __launch_bounds__(64) void k_positives(
    const float* __restrict__ locs,     // (N,P,4)
    const float* __restrict__ scores,   // (N,P,20)
    char* ws) {
    __shared__ float sh_l[2], sh_c[2];
    const int img = blockIdx.x;
    const int cnt = ((const int*)(ws + 16))[img];
    const Ent* ents = (const Ent*)(ws + WS_ENT_OFF);

    float lsum = 0.0f, csum = 0.0f;
    for (int t = threadIdx.x; t < cnt; t += blockDim.x) {
        Ent e = ents[img * ENT_PER_IMG + t];
        int p = e.p;
        int l = (p < 16384) ? 0 : (p < 20480) ? 1 : (p < 21504) ? 2
                                              : (p < 21760) ? 3 : 4;
        int j = p - c_OFF[l];
        int W = c_W[l]; float s = c_S[l];
        float invW = 1.0f / (float)W;
        int y = j / W, x = j - y * W;
        float pcx = ((float)x + 0.5f) * invW;
        float pcy = ((float)y + 0.5f) * invW;

        const float* g = locs + ((long)img * P_TOT + p) * 4;
        float cx = g[0] * s * 0.1f + pcx;
        float cy = g[1] * s * 0.1f + pcy;
        float w  = __expf(g[2] * 0.2f) * s;
        float h  = __expf(g[3] * 0.2f) * s;
        lsum += smooth_l1((cx - 0.5f * w) - e.b0)
              + smooth_l1((cy - 0.5f * h) - e.b1)
              + smooth_l1((cx + 0.5f * w) - e.b2)
              + smooth_l1((cy + 0.5f * h) - e.b3);

        // correction for matched logit: add (-alpha*term1 + (1-alpha)*term2)
        float xl = scores[((long)img * P_TOT + p) * NCLS + (e.lab - 1)];
        float ax = fabsf(xl);
        float l1p = log1pf(__expf(-ax));
        float ls_p = fminf(xl, 0.0f) - l1p;   // log sigmoid(x)
        float ls_n = fminf(-xl, 0.0f) - l1p;  // log sigmoid(-x)
        float pg = __expf(GAMMA_F * ls_p);    // p^gamma
        float qg = __expf(GAMMA_F * ls_n);    // (1-p)^gamma
        csum += -ALPHA_F * qg * ls_p + (1.0f - ALPHA_F) * pg * ls_n;
    }
    // two waves: WMMA per-wave reduction (fully converged -> EXEC all ones)
    float wl = wave_sum_wmma(lsum);
    float wc = wave_sum_wmma(csum);
    int wid = threadIdx.x >> 5;
    if ((threadIdx.x & 31) == 0) { sh_l[wid] = wl; sh_c[wid] = wc; }
    __syncthreads();
    if (threadIdx.x == 0) {
        atomicAdd((float*)ws + 1, sh_l[0] + sh_l[1]);   // loc accumulator
        atomicAdd((float*)ws + 0, sh_c[0] + sh_c[1]);   // conf accumulator
    }
}

// ---------------- K4: combine ----------------
__global__ void k_finalize(const char* ws, float* out) {
    float conf = ((const float*)ws)[0];
    float loc  = ((const float*)ws)[1];
    float np   = (float)((const unsigned*)ws)[3];
    out[0] = conf / np + loc / (4.0f * np);
}

// ---------------- launcher ----------------
extern "C" void kernel_launch(void* const* d_in, const int* in_sizes, int n_in,
                              void* d_out, int out_size, void* d_ws, size_t ws_size,
                              hipStream_t stream) {
    const float* locs   = (const float*)d_in[0];  // (32, 21824, 4)
    const float* scores = (const float*)d_in[1];  // (32, 21824, 20)
    const float* boxes  = (const float*)d_in[2];  // (32, 16, 4)
    const int*   labels = (const int*)d_in[3];    // (32, 16)
    char* ws = (char*)d_ws;
    float* out = (float*)d_out;

    const long total4 = (long)N_IMG * P_TOT * NCLS / 4;  // 3,491,840 float4s

    k_init<<<1, 64, 0, stream>>>(ws);
    k_assign<<<N_IMG, 512, 0, stream>>>(boxes, labels, ws);
    k_focal_base<<<4096, 256, 0, stream>>>(scores, ws, total4);
    k_positives<<<N_IMG, 64, 0, stream>>>(locs, scores, ws);
    k_finalize<<<1, 1, 0, stream>>>(ws, out);
}